// Attention_38869454029278
// MI455X (gfx1250) — compile-verified
//
#include <hip/hip_runtime.h>

typedef float v2f __attribute__((ext_vector_type(2)));
typedef float v4f __attribute__((ext_vector_type(4)));
typedef float v8f __attribute__((ext_vector_type(8)));

#define BATCH 2
#define NHEAD 12
#define SEQ   2048
#define DH    64
#define QT    16          // query rows per workgroup
#define NKT   (SEQ / 16)  // 128 key tiles
#define NWAVE 8           // 256 threads, wave32

// D = A(16x4 f32) * B(4x16 f32) + C(16x16 f32), wave32 WMMA.
__device__ __forceinline__ v8f wmma_f32_4(v2f a, v2f b, v8f c) {
  // 8 args: (neg_a, A, neg_b, B, c_mod, C, reuse_a, reuse_b)
  return __builtin_amdgcn_wmma_f32_16x16x4_f32(false, a, false, b, (short)0, c,
                                               false, false);
}

__global__ void attn_fp32_wmma_kernel(const float* __restrict__ Q,
                                      const float* __restrict__ K,
                                      const float* __restrict__ V,
                                      const int*   __restrict__ mask,
                                      float* __restrict__ out,    // [B,H,S,D]
                                      float* __restrict__ pout)   // [B,H,S,S]
{
  extern __shared__ float smem[];
  float* sS     = smem;                 // QT*SEQ scores -> normalized p
  float* red    = sS + QT * SEQ;        // 256 reduction partials
  float* rowmax = red + 256;            // 16
  float* rowsum = rowmax + 16;          // 16
  float* pvred  = rowsum + 16;          // 4 * 256 PV partial tiles

  const int tid  = threadIdx.x;
  const int lane = tid & 31;            // wave32
  const int wave = tid >> 5;

  const int nqt = SEQ / QT;             // 128 query tiles per (b,h)
  const int bh  = blockIdx.x / nqt;     // 0..23
  const int qt  = blockIdx.x % nqt;
  const int b   = bh / NHEAD;
  const int q0  = qt * QT;

  const int l16    = lane & 15;
  const int hi     = lane >> 4;         // 0: lanes 0-15, 1: lanes 16-31
  const int doff   = hi * 2;            // K-dim pair selector for A/B frags
  const int rowoff = hi * 8;            // C/D row group

  const float* Qb = Q + (size_t)(bh * SEQ + q0) * DH;
  const float* Kb = K + (size_t)bh * SEQ * DH;
  const float* Vb = V + (size_t)bh * SEQ * DH;
  const int*   mb = mask + (size_t)b * SEQ;

  // ---------------- Phase 1: masked scaled scores -> LDS -----------------
  const float* qrow = Qb + (size_t)l16 * DH + doff;   // A frag base (this lane's Q row)
  for (int kt = wave; kt < NKT; kt += NWAVE) {
    const int kbase = kt * 16;
    const int key   = kbase + l16;                    // this lane's key column
    const float* krow = Kb + (size_t)key * DH + doff; // B frag base (K row = col of K^T)
    v8f acc = {0.f, 0.f, 0.f, 0.f, 0.f, 0.f, 0.f, 0.f};
#pragma unroll
    for (int c = 0; c < DH / 4; ++c) {
      v2f a = *(const v2f*)(qrow + 4 * c);
      v2f kk = *(const v2f*)(krow + 4 * c);
      acc = wmma_f32_4(a, kk, acc);
    }
    const bool masked = (mb[key] == 0);
    float* dst = sS + (size_t)rowoff * SEQ + kbase + l16;
#pragma unroll
    for (int r = 0; r < 8; ++r) {
      float sv = acc[r] * 0.125f;                     // 1/sqrt(64)
      dst[(size_t)r * SEQ] = masked ? -1e9f : sv;
    }
  }
  __syncthreads();

  // ---------------- Phase 2: softmax (rowmax, exp, rowsum, normalize) ----
  const int row = tid >> 4;             // 16 threads per query row
  const int sub = tid & 15;
  float* srow = sS + (size_t)row * SEQ;

  float pmax = -3.0e38f;
  for (int i = sub; i < SEQ; i += 16) pmax = fmaxf(pmax, srow[i]);
  red[tid] = pmax;
  __syncthreads();
  if (sub == 0) {
    float m = red[row * 16];
#pragma unroll
    for (int j = 1; j < 16; ++j) m = fmaxf(m, red[row * 16 + j]);
    rowmax[row] = m;
  }
  __syncthreads();

  const float m = rowmax[row];
  float psum = 0.0f;
  for (int i = sub; i < SEQ; i += 16) {
    float e = __expf(srow[i] - m);
    srow[i] = e;
    psum += e;
  }
  red[tid] = psum;
  __syncthreads();
  if (sub == 0) {
    float s = 0.0f;
#pragma unroll
    for (int j = 0; j < 16; ++j) s += red[row * 16 + j];
    rowsum[row] = s;
  }
  __syncthreads();

  // Normalize p in place in LDS (so LDS holds the final p_attn tile and the
  // async LDS->global store below needs no VGPR round-trip).
  {
    const float inv = 1.0f / rowsum[row];
    for (int i = sub; i < SEQ; i += 16) srow[i] *= inv;
  }
  __syncthreads();

  // ---------------- Phase 3: p_attn -> global via async LDS->mem store ---
  // 16 contiguous rows of p_attn exactly mirror the LDS layout: stream the
  // 128KB tile with global_store_async_from_lds_b128 (ASYNCcnt path); the
  // stores drain in the background while PV runs below.
  {
    float* pg = pout + ((size_t)bh * SEQ + q0) * SEQ;
    const int total16B = QT * SEQ / 4;                 // 8192 x 16-byte chunks
    for (int i = tid; i < total16B; i += 256) {
      const unsigned lds_byte = (unsigned)i * 16u;     // dynamic LDS base = 0
      const float* gaddr = pg + (size_t)i * 4;
      asm volatile("global_store_async_from_lds_b128 %0, %1, off"
                   :: "v"(gaddr), "v"(lds_byte)
                   : "memory");
    }
  }

  // ---------------- Phase 4: out = P @ V (P already normalized) ----------
  const int ntile = wave & 3;           // d-tile: d0..d0+15
  const int khalf = wave >> 2;          // split K range across wave pairs
  const int d0 = ntile * 16;
  const float* arow = sS + (size_t)l16 * SEQ + doff;    // P A-frag base
  v8f acc = {0.f, 0.f, 0.f, 0.f, 0.f, 0.f, 0.f, 0.f};
  const int c0 = khalf * (SEQ / 8);     // 256 k-chunks of 4 each
  for (int c = c0; c < c0 + SEQ / 8; ++c) {
    v2f a = *(const v2f*)(arow + 4 * c);                // {P[q][4c+doff], +1}
    const int kr = 4 * c + doff;
    v2f vv;
    vv.x = Vb[(size_t)kr * DH + d0 + l16];
    vv.y = Vb[(size_t)(kr + 1) * DH + d0 + l16];
    acc = wmma_f32_4(a, vv, acc);
  }
  if (khalf == 1) {
    float* pr = pvred + ntile * 256;
#pragma unroll
    for (int r = 0; r < 8; ++r) pr[(rowoff + r) * 16 + l16] = acc[r];
  }
  __syncthreads();
  if (khalf == 0) {
    const float* pr = pvred + ntile * 256;
    float* og = out + ((size_t)bh * SEQ + q0) * DH + d0;
#pragma unroll
    for (int r = 0; r < 8; ++r) {
      const int ql = rowoff + r;
      const float v = acc[r] + pr[ql * 16 + l16];
      __builtin_nontemporal_store(v, &og[(size_t)ql * DH + l16]); // streaming out
    }
  }

  // Drain async p_attn stores before wave retirement (S_ENDPGM also has an
  // implicit wait-idle; this makes the dependency explicit).
  asm volatile("s_wait_asynccnt 0x0" ::: "memory");
}

extern "C" void kernel_launch(void* const* d_in, const int* in_sizes, int n_in,
                              void* d_out, int out_size, void* d_ws, size_t ws_size,
                              hipStream_t stream) {
  (void)in_sizes; (void)n_in; (void)out_size; (void)d_ws; (void)ws_size;
  const float* Q    = (const float*)d_in[0];
  const float* K    = (const float*)d_in[1];
  const float* V    = (const float*)d_in[2];
  const int*   mask = (const int*)d_in[3];

  float* out  = (float*)d_out;                                  // B*H*S*D first
  float* pout = out + (size_t)BATCH * NHEAD * SEQ * DH;         // then B*H*S*S

  const size_t shmem =
      (size_t)(QT * SEQ + 256 + 16 + 16 + 4 * 256) * sizeof(float); // ~136 KB (<320 KB/WGP)
  dim3 grid(BATCH * NHEAD * (SEQ / QT));   // 3072 workgroups
  dim3 block(256);                          // 8 wave32 waves
  attn_fp32_wmma_kernel<<<grid, block, shmem, stream>>>(Q, K, V, mask, out, pout);
}